// PNNLayer_84043920048504
// MI455X (gfx1250) — compile-verified
//
#include <hip/hip_runtime.h>

// CDNA5 / gfx1250, wave32. fp32 WMMA: D(16x16,f32) = A(16x4,f32) x B(4x16,f32) + C
typedef float v2f __attribute__((ext_vector_type(2)));
typedef float v8f __attribute__((ext_vector_type(8)));

#define NN 30000
#define AA 32
#define DD 128
#define NH 1875            // NN/16 half-window count
#define SA_STRIDE 130      // LDS pad for 16x128 A strip (b64 reads, few conflicts)
#define SP_STRIDE 136      // LDS pad for P (32x128): 136%64==8 -> half-waves hit disjoint banks
#define SS_STRIDE 34       // LDS pad for dists strip 16x32

// ---------------- Kernel 1: prep ----------------
// blocks 0..127  : Wt[r][k] = W_hidden[k][128+r]        (transpose of W2)
// blocks 128..159: P[a][k]  = (1/32) * sum_d embeds[id[a]][d] * W_hidden[k][d]
__global__ void __launch_bounds__(128)
prep_kernel(const float* __restrict__ embeds, const float* __restrict__ Wh,
            const int* __restrict__ ids, float* __restrict__ Wt, float* __restrict__ P) {
  int r = blockIdx.x, k = threadIdx.x;
  if (r < DD) {
    Wt[r * DD + k] = Wh[k * (2 * DD) + DD + r];
  } else {
    int a = r - DD;
    const float* e = embeds + (long)ids[a] * DD;
    const float* w = Wh + k * (2 * DD);
    float s = 0.f;
#pragma unroll 8
    for (int d = 0; d < DD; ++d) s = fmaf(e[d], w[d], s);
    P[a * DD + k] = s * (1.0f / 32.0f);
  }
}

// ---------------- Kernel 2: Q = embeds @ W2^T via V_WMMA_F32_16X16X4_F32 ----------------
// grid = 1875 (16-row strips), block = 256 = 8 waves; wave w owns 16x16 tile at cols [16w,16w+16)
__global__ void __launch_bounds__(256)
qgemm_kernel(const float* __restrict__ embeds, const float* __restrict__ Wt,
             float* __restrict__ Q) {
  __shared__ float sA[16 * SA_STRIDE];   // 16 x 128 strip of embeds (row-major, padded)
  const int tid = threadIdx.x;
  const int n0 = blockIdx.x * 16;

  // cooperative stage of A strip: 2048 floats as 1024 float2
  for (int i = tid; i < 1024; i += 256) {
    int r = i >> 6, c2 = i & 63;
    v2f v = *(const v2f*)(embeds + (long)(n0 + r) * DD + c2 * 2);
    *(v2f*)(&sA[r * SA_STRIDE + c2 * 2]) = v;   // 8B-aligned: 130*4 % 8 == 0
  }
  __syncthreads();

  const int lane = tid & 31, wave = tid >> 5;
  const int lr = lane & 15, hi = lane >> 4;
  const int col = wave * 16 + lr;

  v8f acc = {};
#pragma unroll
  for (int k0 = 0; k0 < DD; k0 += 4) {
    const int ka = k0 + hi * 2;                    // A layout: lanes16-31 hold K+2,K+3
    v2f a = *(const v2f*)(&sA[lr * SA_STRIDE + ka]);
    v2f b;                                          // B: VGPR0=row ka, VGPR1=row ka+1
    b.x = Wt[ka * DD + col];
    b.y = Wt[(ka + 1) * DD + col];
    acc = __builtin_amdgcn_wmma_f32_16x16x4_f32(false, a, false, b, (short)0, acc, false, false);
  }
#pragma unroll
  for (int vi = 0; vi < 8; ++vi) {                  // C/D: VGPR v -> M = v (+8 for hi half)
    int row = n0 + vi + hi * 8;
    Q[(long)row * DD + col] = acc[vi];
  }
}

// ---------------- Kernel 3: half-window sums H[i] = sum of 16 Q rows ----------------
__global__ void __launch_bounds__(128)
hsum_kernel(const float* __restrict__ Q, float* __restrict__ H) {
  int i = blockIdx.x, k = threadIdx.x;
  const float* q = Q + (long)i * 16 * DD + k;
  float s = 0.f;
#pragma unroll
  for (int r = 0; r < 16; ++r) s += q[r * DD];
  H[i * DD + k] = s;
}

// ---------------- Kernel 4: out = dists^T @ P' (WMMA, K=32) + windowed self + bias ----------------
__global__ void __launch_bounds__(256)
final_kernel(const float* __restrict__ dists, const float* __restrict__ P,
             const float* __restrict__ H, const float* __restrict__ bias,
             float* __restrict__ out) {
  __shared__ float sS[16 * SS_STRIDE];   // sS[r][a] = dists[a][n0+r]
  __shared__ float sP[AA * SP_STRIDE];   // P'(=P/32), padded
  const int tid = threadIdx.x;
  const int n0 = blockIdx.x * 16;

  {  // stage dists strip: thread t -> (a = t/8, rows 2*(t%8), +1); contiguous-in-n float2 load
    int a = tid >> 3, rr = (tid & 7) * 2;
    v2f v = *(const v2f*)(dists + (long)a * NN + n0 + rr);
    sS[rr * SS_STRIDE + a] = v.x;
    sS[(rr + 1) * SS_STRIDE + a] = v.y;
  }
  for (int i = tid; i < 1024; i += 256) {  // stage P: 32x128 = 1024 float4
    int r = i >> 5, c4 = i & 31;
    float4 v = *(const float4*)(P + r * DD + c4 * 4);
    *(float4*)(&sP[r * SP_STRIDE + c4 * 4]) = v;   // 136*4 % 16 == 0 -> 16B aligned
  }
  __syncthreads();

  const int lane = tid & 31, wave = tid >> 5;
  const int lr = lane & 15, hi = lane >> 4;
  const int col = wave * 16 + lr;

  v8f acc = {};
#pragma unroll
  for (int k0 = 0; k0 < AA; k0 += 4) {
    const int ka = k0 + hi * 2;
    v2f a = *(const v2f*)(&sS[lr * SS_STRIDE + ka]);
    v2f b;
    b.x = sP[ka * SP_STRIDE + col];
    b.y = sP[(ka + 1) * SP_STRIDE + col];
    acc = __builtin_amdgcn_wmma_f32_16x16x4_f32(false, a, false, b, (short)0, acc, false, false);
  }

  const float bv = bias[col];
#pragma unroll
  for (int vi = 0; vi < 8; ++vi) {
    int n = n0 + vi + hi * 8;
    int i1 = (2 * n) % NH;
    int i2 = (i1 + 1 == NH) ? 0 : i1 + 1;
    float selfterm = (H[i1 * DD + col] + H[i2 * DD + col]) * (1.0f / 32.0f);
    out[(long)n * DD + col] = acc[vi] + selfterm + bv;
  }
}

// ---------------- launcher ----------------
extern "C" void kernel_launch(void* const* d_in, const int* in_sizes, int n_in,
                              void* d_out, int out_size, void* d_ws, size_t ws_size,
                              hipStream_t stream) {
  const float* embeds = (const float*)d_in[0];   // (N, D)
  const float* dists  = (const float*)d_in[1];   // (A, N)
  const float* Wh     = (const float*)d_in[2];   // (D, 2D)
  const float* bias   = (const float*)d_in[3];   // (D,)
  const int*   ids    = (const int*)d_in[4];     // (A,)
  float* out = (float*)d_out;                    // (N, D) — also reused as Q scratch
  float* ws  = (float*)d_ws;

  float* Wt = ws;                          // 128*128 floats  (W2 transposed)
  float* P  = Wt + DD * DD;                // 32*128 floats   (P/32)
  float* H  = P + AA * DD;                 // 1875*128 floats (half-window sums)

  prep_kernel <<<DD + AA, DD, 0, stream>>>(embeds, Wh, ids, Wt, P);
  qgemm_kernel<<<NH, 256, 0, stream>>>(embeds, Wt, out);   // out <- Q (scratch)
  hsum_kernel <<<NH, DD, 0, stream>>>(out, H);
  final_kernel<<<NH, 256, 0, stream>>>(dists, P, H, bias, out);  // overwrite with result
}